// ExpertsChooseMaskedExpand_64080912056708
// MI455X (gfx1250) — compile-verified
//
#include <hip/hip_runtime.h>

// MoE experts-choose on gfx1250:
//   xd = dm^T @ x   (per b,e)            -> workspace
//   z  = ca @ xd    (per b,e, reassoc.)  -> workspace
//   out = z @ W2^T + bias * rowsum(ca)   (per b)
// fp32 WMMA (V_WMMA_F32_16X16X4_F32), async global->LDS (ASYNCcnt) staging,
// TRIPLE-buffered K pipeline with partial s_wait_asynccnt,
// 128x128 block tiles / 32x64 wave tiles, contiguous b64 LDS frag loads.

typedef __attribute__((ext_vector_type(2))) float v2f;
typedef __attribute__((ext_vector_type(4))) float f32x4;
typedef __attribute__((ext_vector_type(8))) float v8f;
typedef __attribute__((ext_vector_type(4))) int v4i;

#if __has_builtin(__builtin_amdgcn_global_load_async_to_lds_b128)
#define HAVE_ASYNC_CP 1
#else
#define HAVE_ASYNC_CP 0
#endif
#if __has_builtin(__builtin_amdgcn_s_wait_asynccnt)
#define HAVE_ASYNC_WAIT 1
#else
#define HAVE_ASYNC_WAIT 0
#endif

typedef __attribute__((address_space(1))) v4i* gv4_t;
typedef __attribute__((address_space(3))) v4i* lv4_t;

// Copy 16 contiguous bytes global -> LDS. Async (ASYNCcnt) when available.
__device__ __forceinline__ void cp16(void* lds, const void* gmem) {
#if HAVE_ASYNC_CP
  __builtin_amdgcn_global_load_async_to_lds_b128((gv4_t)gmem, (lv4_t)lds, 0, 0);
#else
  *(f32x4*)lds = *(const f32x4*)gmem;
#endif
}

// Each stage issues 4 cp16 per thread. Loads complete in order, so
// asynccnt<=4 retires the oldest stage while the next keeps streaming.
__device__ __forceinline__ void wait_async_tail() {
#if HAVE_ASYNC_CP && HAVE_ASYNC_WAIT
  __builtin_amdgcn_s_wait_asynccnt(4);
#endif
  __syncthreads();
}
__device__ __forceinline__ void wait_async_all() {
#if HAVE_ASYNC_CP && HAVE_ASYNC_WAIT
  __builtin_amdgcn_s_wait_asynccnt(0);
#endif
  __syncthreads();
}

__device__ __forceinline__ v8f wmma4(v2f a, v2f b, v8f c) {
  return __builtin_amdgcn_wmma_f32_16x16x4_f32(false, a, false, b, (short)0, c,
                                               false, false);
}

static constexpr int Bb = 4, Tt = 1024, Ee = 4, Cc = 512, Ii = 512;
static constexpr int IN = 2048, OUT = 8192;

// ---------------------------------------------------------------------------
// s[b,t] = sum_{e,c} ca[b,t,e,c]     (one wave per row of 2048)
// ---------------------------------------------------------------------------
__global__ void __launch_bounds__(128)
moe_rowsum_s(const float* __restrict__ ca, float* __restrict__ s) {
  const int row  = blockIdx.x * 4 + (threadIdx.x >> 5);
  const int lane = threadIdx.x & 31;
  const float* p = ca + (size_t)row * (Ee * Cc);
  float acc = 0.f;
  #pragma unroll 4
  for (int i = lane; i < Ee * Cc; i += 32) acc += p[i];
  #pragma unroll
  for (int off = 16; off > 0; off >>= 1) acc += __shfl_down(acc, off, 32);
  if (lane == 0) s[row] = acc;
}

// ---------------------------------------------------------------------------
// Kernel 1: xd[b,e,c,i] = sum_t dm[b,t,e,c] * x[b,t,e*512+i]
//   A is K-major in memory (c contiguous), B row-major: both tiles [k][row]
// ---------------------------------------------------------------------------
__global__ void __launch_bounds__(256)
moe_dispatch_xd(const float* __restrict__ dm, const float* __restrict__ x,
                float* __restrict__ xd) {
  __shared__ float Ak[3][16][136];  // [k][m], pitch 136: halves 16 banks apart
  __shared__ float Bs[3][16][136];  // [k][n]
  const int be = blockIdx.z, b = be >> 2, e = be & 3;
  const int m0 = blockIdx.x * 128, n0 = blockIdx.y * 128;
  const float* Ab = dm + (size_t)b * Tt * Ee * Cc + (size_t)e * Cc; // A[c][t]=Ab[t*2048+c]
  const float* Bp = x + (size_t)b * Tt * IN + (size_t)e * Ii;       // B[t][i]=Bp[t*IN+i]
  const int tid = threadIdx.x, lane = tid & 31, wave = tid >> 5;
  const int wm = (wave & 3) * 32, wn = (wave >> 2) * 64;
  const int lm = lane & 15, kh = (lane >> 4) * 2, mh = (lane >> 4) * 8;

  auto issue = [&](int p, int k0) {
    #pragma unroll
    for (int it = 0; it < 2; ++it) {
      int c = tid + it * 256, r = c >> 5, j = (c & 31) * 4;
      cp16(&Ak[p][r][j], Ab + (size_t)(k0 + r) * (Ee * Cc) + (m0 + j));
      cp16(&Bs[p][r][j], Bp + (size_t)(k0 + r) * IN + (n0 + j));
    }
  };

  v8f acc[2][4] = {};
  int cur = 0, nxt = 2;
  issue(0, 0);
  issue(1, 16);
  for (int k0 = 0; k0 < Tt; k0 += 16) {
    if (k0 + 16 < Tt) wait_async_tail(); else wait_async_all();
    if (k0 + 32 < Tt) issue(nxt, k0 + 32);
    #pragma unroll
    for (int kk = 0; kk < 16; kk += 4) {
      v2f a[2], bf[4];
      #pragma unroll
      for (int mt = 0; mt < 2; ++mt) {
        a[mt].x = Ak[cur][kk + kh][wm + mt * 16 + lm];
        a[mt].y = Ak[cur][kk + kh + 1][wm + mt * 16 + lm];
      }
      #pragma unroll
      for (int nt = 0; nt < 4; ++nt) {
        bf[nt].x = Bs[cur][kk + kh][wn + nt * 16 + lm];
        bf[nt].y = Bs[cur][kk + kh + 1][wn + nt * 16 + lm];
      }
      #pragma unroll
      for (int mt = 0; mt < 2; ++mt)
        #pragma unroll
        for (int nt = 0; nt < 4; ++nt)
          acc[mt][nt] = wmma4(a[mt], bf[nt], acc[mt][nt]);
    }
    cur = (cur == 2) ? 0 : cur + 1;
    nxt = (nxt == 2) ? 0 : nxt + 1;
  }
  float* Ob = xd + (size_t)be * Cc * Ii;
  #pragma unroll
  for (int mt = 0; mt < 2; ++mt)
    #pragma unroll
    for (int nt = 0; nt < 4; ++nt)
      #pragma unroll
      for (int r = 0; r < 8; ++r) {
        int m = m0 + wm + mt * 16 + mh + r;
        int n = n0 + wn + nt * 16 + lm;
        Ob[(size_t)m * Ii + n] = acc[mt][nt][r];
      }
}

// ---------------------------------------------------------------------------
// Kernel 2: zc[b,t,e*512+i] = sum_c ca[b,t,e,c] * xd[b,e,c,i]
//   A row-major (k contiguous) -> [m][k] pitch 20 (b64 frags); B -> [k][n]
// ---------------------------------------------------------------------------
__global__ void __launch_bounds__(256)
moe_combine_z(const float* __restrict__ ca, const float* __restrict__ xd,
              float* __restrict__ zc) {
  __shared__ float As[3][128][20];  // [m][k], pitch 20: 20*lm mod 64 distinct
  __shared__ float Bs[3][16][136];  // [k][n]
  const int be = blockIdx.z, b = be >> 2, e = be & 3;
  const int m0 = blockIdx.x * 128, n0 = blockIdx.y * 128;
  const float* Ab = ca + (size_t)b * Tt * Ee * Cc + (size_t)e * Cc; // A[t][c]=Ab[t*2048+c]
  const float* Bp = xd + (size_t)be * Cc * Ii;                      // B[c][i]
  const int tid = threadIdx.x, lane = tid & 31, wave = tid >> 5;
  const int wm = (wave & 3) * 32, wn = (wave >> 2) * 64;
  const int lm = lane & 15, kh = (lane >> 4) * 2, mh = (lane >> 4) * 8;

  auto issue = [&](int p, int k0) {
    #pragma unroll
    for (int it = 0; it < 2; ++it) {
      int c = tid + it * 256;
      int ra = c >> 2, ja = (c & 3) * 4;
      cp16(&As[p][ra][ja], Ab + (size_t)(m0 + ra) * (Ee * Cc) + (k0 + ja));
      int rb = c >> 5, jb = (c & 31) * 4;
      cp16(&Bs[p][rb][jb], Bp + (size_t)(k0 + rb) * Ii + (n0 + jb));
    }
  };

  v8f acc[2][4] = {};
  int cur = 0, nxt = 2;
  issue(0, 0);
  issue(1, 16);
  for (int k0 = 0; k0 < Cc; k0 += 16) {
    if (k0 + 16 < Cc) wait_async_tail(); else wait_async_all();
    if (k0 + 32 < Cc) issue(nxt, k0 + 32);
    #pragma unroll
    for (int kk = 0; kk < 16; kk += 4) {
      v2f a[2], bf[4];
      #pragma unroll
      for (int mt = 0; mt < 2; ++mt)
        a[mt] = *(const v2f*)&As[cur][wm + mt * 16 + lm][kk + kh];
      #pragma unroll
      for (int nt = 0; nt < 4; ++nt) {
        bf[nt].x = Bs[cur][kk + kh][wn + nt * 16 + lm];
        bf[nt].y = Bs[cur][kk + kh + 1][wn + nt * 16 + lm];
      }
      #pragma unroll
      for (int mt = 0; mt < 2; ++mt)
        #pragma unroll
        for (int nt = 0; nt < 4; ++nt)
          acc[mt][nt] = wmma4(a[mt], bf[nt], acc[mt][nt]);
    }
    cur = (cur == 2) ? 0 : cur + 1;
    nxt = (nxt == 2) ? 0 : nxt + 1;
  }
  #pragma unroll
  for (int mt = 0; mt < 2; ++mt)
    #pragma unroll
    for (int nt = 0; nt < 4; ++nt)
      #pragma unroll
      for (int r = 0; r < 8; ++r) {
        int m = m0 + wm + mt * 16 + mh + r;
        int n = n0 + wn + nt * 16 + lm;
        zc[((size_t)b * Tt + m) * IN + (size_t)e * Ii + n] = acc[mt][nt][r];
      }
}

// ---------------------------------------------------------------------------
// Kernel 3: out[b,t,o] = sum_k zc[b,t,k]*W2[o,k] + bias[o]*s[b,t]
//   W2[o, e*512+ii] = weight_flat[e*OUT*512 + o*512 + ii]
//   A -> [m][k] pitch 20, B -> [n][k] pitch 20: BOTH frags contiguous b64
// ---------------------------------------------------------------------------
__global__ void __launch_bounds__(256)
moe_final_gemm(const float* __restrict__ zc, const float* __restrict__ w,
               const float* __restrict__ bias, const float* __restrict__ s,
               float* __restrict__ out) {
  __shared__ float As[3][128][20];  // [m][k]
  __shared__ float Bt[3][128][20];  // [n][k]
  const int b = blockIdx.z;
  const int m0 = blockIdx.x * 128, n0 = blockIdx.y * 128;
  const float* Ab = zc + (size_t)b * Tt * IN;  // A[t][k]
  const int tid = threadIdx.x, lane = tid & 31, wave = tid >> 5;
  const int wm = (wave & 3) * 32, wn = (wave >> 2) * 64;
  const int lm = lane & 15, kh = (lane >> 4) * 2, mh = (lane >> 4) * 8;

  auto issue = [&](int p, int k0) {
    // expert block of the reshaped weight; 16-wide k-tiles never straddle experts
    const float* Wb = w + (size_t)(k0 >> 9) * ((size_t)OUT * Ii) + (k0 & 511);
    #pragma unroll
    for (int it = 0; it < 2; ++it) {
      int c = tid + it * 256, r = c >> 2, j = (c & 3) * 4;
      cp16(&As[p][r][j], Ab + (size_t)(m0 + r) * IN + (k0 + j));
      cp16(&Bt[p][r][j], Wb + (size_t)(n0 + r) * Ii + j);
    }
  };

  v8f acc[2][4] = {};
  int cur = 0, nxt = 2;
  issue(0, 0);
  issue(1, 16);
  for (int k0 = 0; k0 < IN; k0 += 16) {
    if (k0 + 16 < IN) wait_async_tail(); else wait_async_all();
    if (k0 + 32 < IN) issue(nxt, k0 + 32);
    #pragma unroll
    for (int kk = 0; kk < 16; kk += 4) {
      v2f a[2], bf[4];
      #pragma unroll
      for (int mt = 0; mt < 2; ++mt)
        a[mt] = *(const v2f*)&As[cur][wm + mt * 16 + lm][kk + kh];
      #pragma unroll
      for (int nt = 0; nt < 4; ++nt)
        bf[nt] = *(const v2f*)&Bt[cur][wn + nt * 16 + lm][kk + kh];
      #pragma unroll
      for (int mt = 0; mt < 2; ++mt)
        #pragma unroll
        for (int nt = 0; nt < 4; ++nt)
          acc[mt][nt] = wmma4(a[mt], bf[nt], acc[mt][nt]);
    }
    cur = (cur == 2) ? 0 : cur + 1;
    nxt = (nxt == 2) ? 0 : nxt + 1;
  }
  // epilogue: + bias[o] * s[b,t]
  float sv[2][8];
  #pragma unroll
  for (int mt = 0; mt < 2; ++mt)
    #pragma unroll
    for (int r = 0; r < 8; ++r)
      sv[mt][r] = s[(size_t)b * Tt + m0 + wm + mt * 16 + mh + r];
  float bn[4];
  #pragma unroll
  for (int nt = 0; nt < 4; ++nt) bn[nt] = bias[n0 + wn + nt * 16 + lm];
  #pragma unroll
  for (int mt = 0; mt < 2; ++mt)
    #pragma unroll
    for (int nt = 0; nt < 4; ++nt)
      #pragma unroll
      for (int r = 0; r < 8; ++r) {
        int m = m0 + wm + mt * 16 + mh + r;
        int n = n0 + wn + nt * 16 + lm;
        out[((size_t)b * Tt + m) * OUT + n] = acc[mt][nt][r] + bn[nt] * sv[mt][r];
      }
}

// ---------------------------------------------------------------------------
extern "C" void kernel_launch(void* const* d_in, const int* in_sizes, int n_in,
                              void* d_out, int out_size, void* d_ws, size_t ws_size,
                              hipStream_t stream) {
  const float* x    = (const float*)d_in[0];  // (B,T,IN)
  const float* ca   = (const float*)d_in[1];  // (B,T,E,C)
  const float* dm   = (const float*)d_in[2];  // (B,T,E,C)
  const float* w    = (const float*)d_in[3];  // (OUT,IN) flat
  const float* bias = (const float*)d_in[4];  // (OUT,)
  float* out = (float*)d_out;

  float* ws = (float*)d_ws;
  float* xd = ws;                              // 16 MiB
  float* zc = xd + (size_t)Bb * Ee * Cc * Ii;  // 32 MiB
  float* s  = zc + (size_t)Bb * Tt * IN;       // 16 KiB

  moe_rowsum_s   <<<dim3(Bb * Tt / 4), dim3(128), 0, stream>>>(ca, s);
  moe_dispatch_xd<<<dim3(Cc / 128, Ii / 128, Bb * Ee), dim3(256), 0, stream>>>(dm, x, xd);
  moe_combine_z  <<<dim3(Tt / 128, Ii / 128, Bb * Ee), dim3(256), 0, stream>>>(ca, xd, zc);
  moe_final_gemm <<<dim3(Tt / 128, OUT / 128, Bb), dim3(256), 0, stream>>>(zc, w, bias, s, out);
}